// RecurrentPPOPolicy_74663711473843
// MI455X (gfx1250) — compile-verified
//
#include <hip/hip_runtime.h>

// ---------------------------------------------------------------------------
// GAE backward scan, chunked affine-scan decomposition for MI455X (gfx1250).
//
//   delta[t] = r[t] + GAMMA * v[t+1] * (1 - done[t+1]) - v[t]
//   adv[t]   = delta[t] + (GAMMA*LAMBDA) * (1 - done[t+1]) * adv[t+1]
//   ret[t]   = adv[t] + v[t]
//
// Pass 1: per (chunk, env4) affine partials  adv_start = A + P * adv_in
// Pass 2: per env cross-chunk scan, stores each chunk's incoming carry
// Pass 3: per (chunk, env4) final sweep writing adv + returns
// ---------------------------------------------------------------------------

namespace {
constexpr int   T_DIM     = 2048;
constexpr int   N_DIM     = 4096;
constexpr int   NCHUNK    = 64;                 // chunks along T
constexpr int   CHUNK_LEN = T_DIM / NCHUNK;     // 32
constexpr int   NV        = N_DIM / 4;          // 1024 float4 env-groups
constexpr float GAMMA_F   = 0.99f;
constexpr float CGL       = (float)(0.99 * 0.95);  // matches Python double fold
}

// One recurrence step for the affine partial: (a,p) <- (delta + c*a, c*p)
__device__ __forceinline__ void gae_step(float r, float v, float d_next,
                                         float v_next, float& a, float& p) {
  const float nt    = 1.0f - d_next;
  const float delta = r + (GAMMA_F * v_next) * nt - v;
  const float c     = CGL * nt;
  a = delta + c * a;
  p = c * p;
}

// One recurrence step with a concrete carry x = adv[t+1] -> adv[t]
__device__ __forceinline__ float gae_adv(float r, float v, float d_next,
                                         float v_next, float x) {
  const float nt    = 1.0f - d_next;
  const float delta = r + (GAMMA_F * v_next) * nt - v;
  return delta + (CGL * nt) * x;
}

// ---------------------------------------------------------------------------
// Pass 1: chunk affine partials.  grid = NCHUNK * NV threads.
// ---------------------------------------------------------------------------
__global__ __launch_bounds__(256) void gae_partials(
    const float4* __restrict__ r4, const float4* __restrict__ v4,
    const float4* __restrict__ d4,
    float4* __restrict__ A4, float4* __restrict__ P4) {
  const int tid = blockIdx.x * blockDim.x + threadIdx.x;
  const int j   = tid / NV;          // chunk index
  const int g   = tid - j * NV;      // float4 env group
  const int s   = j * CHUNK_LEN;
  const int e   = s + CHUNK_LEN - 1;

  float4 vn, dn;
  if (e == T_DIM - 1) {              // bootstrap: next_value = 0, next_done = 1
    vn = make_float4(0.f, 0.f, 0.f, 0.f);
    dn = make_float4(1.f, 1.f, 1.f, 1.f);
  } else {
    vn = v4[(e + 1) * NV + g];
    dn = d4[(e + 1) * NV + g];
  }

  float4 a = make_float4(0.f, 0.f, 0.f, 0.f);
  float4 p = make_float4(1.f, 1.f, 1.f, 1.f);

#pragma unroll 4
  for (int t = e; t >= s; --t) {
    const int idx = t * NV + g;
    if (t > s) {                     // backward-stride prefetch (global_prefetch_b8)
      __builtin_prefetch((const void*)(r4 + idx - NV), 0, 1);
      __builtin_prefetch((const void*)(v4 + idx - NV), 0, 1);
      __builtin_prefetch((const void*)(d4 + idx - NV), 0, 1);
    }
    const float4 r = r4[idx];
    const float4 v = v4[idx];
    const float4 d = d4[idx];
    gae_step(r.x, v.x, dn.x, vn.x, a.x, p.x);
    gae_step(r.y, v.y, dn.y, vn.y, a.y, p.y);
    gae_step(r.z, v.z, dn.z, vn.z, a.z, p.z);
    gae_step(r.w, v.w, dn.w, vn.w, a.w, p.w);
    vn = v;
    dn = d;
  }
  A4[j * NV + g] = a;
  P4[j * NV + g] = p;
}

// ---------------------------------------------------------------------------
// Pass 2: cross-chunk scan per env (N_DIM threads, 64 steps each).
// Overwrites A[j][n] with the INCOMING carry for chunk j (adv at t=(j+1)*L).
// ---------------------------------------------------------------------------
__global__ __launch_bounds__(256) void gae_chunk_scan(
    float* __restrict__ A, const float* __restrict__ P) {
  const int n = blockIdx.x * blockDim.x + threadIdx.x;   // env index
  float x = 0.f;                                         // carry beyond last chunk
#pragma unroll 8
  for (int j = NCHUNK - 1; j >= 0; --j) {
    const int   idx = j * N_DIM + n;
    const float a   = A[idx];
    const float p   = P[idx];
    A[idx] = x;              // incoming carry for chunk j
    x = a + p * x;           // adv at chunk-j start = carry for chunk j-1
  }
}

// ---------------------------------------------------------------------------
// Pass 3: final sweep, writes advantages and returns.  grid = NCHUNK * NV.
// ---------------------------------------------------------------------------
__global__ __launch_bounds__(256) void gae_finalize(
    const float4* __restrict__ r4, const float4* __restrict__ v4,
    const float4* __restrict__ d4, const float4* __restrict__ X4,
    float4* __restrict__ adv4, float4* __restrict__ ret4) {
  const int tid = blockIdx.x * blockDim.x + threadIdx.x;
  const int j   = tid / NV;
  const int g   = tid - j * NV;
  const int s   = j * CHUNK_LEN;
  const int e   = s + CHUNK_LEN - 1;

  float4 vn, dn;
  if (e == T_DIM - 1) {
    vn = make_float4(0.f, 0.f, 0.f, 0.f);
    dn = make_float4(1.f, 1.f, 1.f, 1.f);
  } else {
    vn = v4[(e + 1) * NV + g];
    dn = d4[(e + 1) * NV + g];
  }

  float4 x = X4[j * NV + g];         // adv carry entering this chunk's tail

#pragma unroll 4
  for (int t = e; t >= s; --t) {
    const int idx = t * NV + g;
    if (t > s) {
      __builtin_prefetch((const void*)(r4 + idx - NV), 0, 1);
      __builtin_prefetch((const void*)(v4 + idx - NV), 0, 1);
      __builtin_prefetch((const void*)(d4 + idx - NV), 0, 1);
    }
    const float4 r = r4[idx];
    const float4 v = v4[idx];
    const float4 d = d4[idx];
    x.x = gae_adv(r.x, v.x, dn.x, vn.x, x.x);
    x.y = gae_adv(r.y, v.y, dn.y, vn.y, x.y);
    x.z = gae_adv(r.z, v.z, dn.z, vn.z, x.z);
    x.w = gae_adv(r.w, v.w, dn.w, vn.w, x.w);
    adv4[idx] = x;
    ret4[idx] = make_float4(x.x + v.x, x.y + v.y, x.z + v.z, x.w + v.w);
    vn = v;
    dn = d;
  }
}

// ---------------------------------------------------------------------------
// Fallback (if workspace is too small): plain per-env backward scan.
// ---------------------------------------------------------------------------
__global__ __launch_bounds__(256) void gae_sequential(
    const float4* __restrict__ r4, const float4* __restrict__ v4,
    const float4* __restrict__ d4,
    float4* __restrict__ adv4, float4* __restrict__ ret4) {
  const int g = blockIdx.x * blockDim.x + threadIdx.x;   // 0..NV-1
  float4 vn = make_float4(0.f, 0.f, 0.f, 0.f);
  float4 dn = make_float4(1.f, 1.f, 1.f, 1.f);
  float4 x  = make_float4(0.f, 0.f, 0.f, 0.f);
#pragma unroll 4
  for (int t = T_DIM - 1; t >= 0; --t) {
    const int idx = t * NV + g;
    if (t > 0) {
      __builtin_prefetch((const void*)(r4 + idx - NV), 0, 1);
      __builtin_prefetch((const void*)(v4 + idx - NV), 0, 1);
      __builtin_prefetch((const void*)(d4 + idx - NV), 0, 1);
    }
    const float4 r = r4[idx];
    const float4 v = v4[idx];
    const float4 d = d4[idx];
    x.x = gae_adv(r.x, v.x, dn.x, vn.x, x.x);
    x.y = gae_adv(r.y, v.y, dn.y, vn.y, x.y);
    x.z = gae_adv(r.z, v.z, dn.z, vn.z, x.z);
    x.w = gae_adv(r.w, v.w, dn.w, vn.w, x.w);
    adv4[idx] = x;
    ret4[idx] = make_float4(x.x + v.x, x.y + v.y, x.z + v.z, x.w + v.w);
    vn = v;
    dn = d;
  }
}

// ---------------------------------------------------------------------------
extern "C" void kernel_launch(void* const* d_in, const int* in_sizes, int n_in,
                              void* d_out, int out_size, void* d_ws, size_t ws_size,
                              hipStream_t stream) {
  (void)in_sizes; (void)n_in; (void)out_size;

  const float4* r4 = (const float4*)d_in[0];   // rewards [T, N] f32
  const float4* v4 = (const float4*)d_in[1];   // values  [T, N] f32
  const float4* d4 = (const float4*)d_in[2];   // dones   [T, N] f32

  float*  out  = (float*)d_out;
  float4* adv4 = (float4*)out;                                   // advantages
  float4* ret4 = (float4*)(out + (size_t)T_DIM * N_DIM);         // returns

  const size_t ws_needed = 2ull * NCHUNK * N_DIM * sizeof(float);  // A + P (2 MB)

  if (d_ws != nullptr && ws_size >= ws_needed) {
    float* A = (float*)d_ws;
    float* P = A + (size_t)NCHUNK * N_DIM;

    const int threads = 256;
    const int blocks_main = (NCHUNK * NV) / threads;   // 256 blocks, 2048 waves

    gae_partials<<<blocks_main, threads, 0, stream>>>(r4, v4, d4,
                                                      (float4*)A, (float4*)P);
    gae_chunk_scan<<<N_DIM / threads, threads, 0, stream>>>(A, P);
    gae_finalize<<<blocks_main, threads, 0, stream>>>(r4, v4, d4,
                                                      (const float4*)A,
                                                      adv4, ret4);
  } else {
    gae_sequential<<<NV / 256, 256, 0, stream>>>(r4, v4, d4, adv4, ret4);
  }
}